// GAT_36361193128616
// MI455X (gfx1250) — compile-verified
//
#include <hip/hip_runtime.h>
#include <cstdint>

typedef __attribute__((ext_vector_type(2))) float v2f;
typedef __attribute__((ext_vector_type(8))) float v8f;

#define NEG_SLOPE 0.2f

// ---------------- helpers ----------------

__device__ __forceinline__ unsigned fkey(float f) {
  unsigned u = __float_as_uint(f);
  return (u & 0x80000000u) ? ~u : (u | 0x80000000u);
}
__device__ __forceinline__ float fdecode(unsigned k) {
  unsigned u = (k & 0x80000000u) ? (k & 0x7FFFFFFFu) : ~k;
  return __uint_as_float(u);
}
__device__ __forceinline__ float lrelu(float x) {
  return x > 0.0f ? x : NEG_SLOPE * x;
}

__global__ void fill_u32_kernel(uint32_t* __restrict__ p, uint32_t val, long long n) {
  long long i = (long long)blockIdx.x * blockDim.x + threadIdx.x;
  if (i < n) p[i] = val;
}

// ---------------- fp32 GEMM via V_WMMA_F32_16X16X4_F32 ----------------
// C[M, NOUT] = A[M, 128] * B[128, NOUT].  One 16-row M-tile per block,
// NOUT/16 waves per block, one 16x16 N-tile per wave.  M must be /16.
template <int NOUT>
__global__ void gemm_f32_wmma(const float* __restrict__ A,
                              const float* __restrict__ B,
                              float* __restrict__ C) {
  constexpr int K   = 128;
  constexpr int NT  = NOUT / 16;
  constexpr int LDA = K + 4;            // pad: 4-float skew -> conflict-free LDS reads
  __shared__ float Alds[16 * LDA];

  const int lane = threadIdx.x & 31;
  const int wave = threadIdx.x >> 5;
  const long long row0 = (long long)blockIdx.x * 16;

  // cooperative A-tile load: 16 rows x 128 floats, float4 (GLOBAL_LOAD_B128)
  for (int idx = threadIdx.x; idx < 16 * (K / 4); idx += 32 * NT) {
    int r  = idx >> 5;                  // 32 float4 per row
    int c4 = idx & 31;
    float4 v = ((const float4*)(A + (row0 + r) * K))[c4];
    float* d = &Alds[r * LDA + c4 * 4];
    d[0] = v.x; d[1] = v.y; d[2] = v.z; d[3] = v.w;
  }
  __syncthreads();

  const int n0 = wave * 16;
  const int ml = lane & 15;             // A: lane -> M row, C/D: lane -> N col
  const int kh = lane >> 4;             // which K-pair half (K+0/1 vs K+2/3)

  v8f acc = {};
#pragma unroll
  for (int k0 = 0; k0 < K; k0 += 4) {
    v2f a, b;
    // A 16x4 layout: lanes0-15 hold {K=k0,k0+1}, lanes16-31 {k0+2,k0+3}, lane&15 = M
    a.x = Alds[ml * LDA + k0 + 2 * kh + 0];
    a.y = Alds[ml * LDA + k0 + 2 * kh + 1];
    // B 4x16 layout: VGPR v, lane-half h -> row K = k0 + v + 2*h, col N = lane&15
    const float* bp = B + (long long)(k0 + 2 * kh) * NOUT + n0 + ml;
    b.x = bp[0];
    b.y = bp[NOUT];
    acc = __builtin_amdgcn_wmma_f32_16x16x4_f32(
        /*neg_a=*/false, a, /*neg_b=*/false, b,
        /*c_mod=*/(short)0, acc, /*reuse_a=*/false, /*reuse_b=*/false);
  }

  // C/D 16x16 f32 layout: VGPR v -> M = v + 8*(lane>=16), N = lane&15
#pragma unroll
  for (int v = 0; v < 8; ++v) {
    long long m = row0 + v + 8 * kh;
    C[m * NOUT + n0 + ml] = acc[v];
  }
}

// ---------------- per-node attention dots ----------------
// aS[n,h] = dot(hfeat[n,h,:], attS[h,:]); one wave per node.
template <int HC, int H>
__global__ void attn_dots(const float* __restrict__ hfeat,
                          const float* __restrict__ attS,
                          const float* __restrict__ attD,
                          float* __restrict__ aS, float* __restrict__ aD, int N) {
  const int lane = threadIdx.x & 31;
  const int wave = threadIdx.x >> 5;
  const int n = blockIdx.x * 8 + wave;        // blockDim.x == 256
  if (n >= N) return;
  constexpr int P = HC / 32;                  // floats per lane
  const float* hp = hfeat + (long long)n * HC + lane * P;
  float s = 0.0f, d = 0.0f;
#pragma unroll
  for (int j = 0; j < P; ++j) {
    float v = hp[j];
    s += v * attS[lane * P + j];
    d += v * attD[lane * P + j];
  }
  constexpr int LPH = 32 / H;                 // lanes per head
#pragma unroll
  for (int off = 1; off < LPH; off <<= 1) {
    s += __shfl_xor(s, off, 32);
    d += __shfl_xor(d, off, 32);
  }
  if ((lane & (LPH - 1)) == 0) {
    int hh = lane / LPH;
    aS[(long long)n * H + hh] = s;
    aD[(long long)n * H + hh] = d;
  }
}

// ---------------- edge phase ----------------
template <int H>
__global__ void edge_max(const int* __restrict__ srcI, const int* __restrict__ dstI,
                         int E, int E2,
                         const float* __restrict__ aS, const float* __restrict__ aD,
                         unsigned* __restrict__ mkey) {
  int i = blockIdx.x * blockDim.x + threadIdx.x;
  if (i >= E2) return;
  int s, d;
  if (i < E) { s = srcI[i]; d = dstI[i]; } else { s = d = i - E; }
#pragma unroll
  for (int hh = 0; hh < H; ++hh) {
    float e = lrelu(aS[(long long)s * H + hh] + aD[(long long)d * H + hh]);
    atomicMax(&mkey[(long long)d * H + hh], fkey(e));
  }
}

template <int H>
__global__ void edge_exp(const int* __restrict__ srcI, const int* __restrict__ dstI,
                         int E, int E2,
                         const float* __restrict__ aS, const float* __restrict__ aD,
                         const unsigned* __restrict__ mkey,
                         float* __restrict__ ex, float* __restrict__ denom) {
  int i = blockIdx.x * blockDim.x + threadIdx.x;
  if (i >= E2) return;
  int s, d;
  if (i < E) { s = srcI[i]; d = dstI[i]; } else { s = d = i - E; }
#pragma unroll
  for (int hh = 0; hh < H; ++hh) {
    float e = lrelu(aS[(long long)s * H + hh] + aD[(long long)d * H + hh]);
    float m = fdecode(mkey[(long long)d * H + hh]);
    float v = expf(e - m);
    ex[(long long)i * H + hh] = v;
    atomicAdd(&denom[(long long)d * H + hh], v);
  }
}

// one thread per (edge, channel): coalesced gather of h[src], f32 atomic scatter
template <int HC, int H>
__global__ void edge_scatter(const int* __restrict__ srcI, const int* __restrict__ dstI,
                             int E, int E2,
                             const float* __restrict__ hfeat,
                             const float* __restrict__ ex,
                             const float* __restrict__ denom,
                             float* __restrict__ out) {
  long long t = (long long)blockIdx.x * blockDim.x + threadIdx.x;
  long long total = (long long)E2 * HC;
  if (t >= total) return;
  int i = (int)(t / HC);
  int c = (int)(t % HC);
  int s, d;
  if (i < E) { s = srcI[i]; d = dstI[i]; } else { s = d = i - E; }
  int hh = c / (HC / H);
  float alpha = ex[(long long)i * H + hh] / denom[(long long)d * H + hh];
  atomicAdd(&out[(long long)d * HC + c], hfeat[(long long)s * HC + c] * alpha);
}

// ---------------- epilogues ----------------
__global__ void bias_elu_kernel(const float* __restrict__ acc, const float* __restrict__ bias,
                                float* __restrict__ out, long long n, int HC) {
  long long t = (long long)blockIdx.x * blockDim.x + threadIdx.x;
  if (t >= n) return;
  float v = acc[t] + bias[t % HC];
  out[t] = v > 0.0f ? v : (expf(v) - 1.0f);   // ELU, alpha=1
}

__global__ void bias_add_kernel(float* __restrict__ out, const float* __restrict__ bias,
                                long long n, int HC) {
  long long t = (long long)blockIdx.x * blockDim.x + threadIdx.x;
  if (t >= n) return;
  out[t] += bias[t % HC];
}

// ---------------- host launch ----------------
static inline unsigned nblk(long long n, int b) { return (unsigned)((n + b - 1) / b); }

extern "C" void kernel_launch(void* const* d_in, const int* in_sizes, int n_in,
                              void* d_out, int out_size, void* d_ws, size_t ws_size,
                              hipStream_t stream) {
  (void)n_in; (void)out_size; (void)ws_size;
  const float* x   = (const float*)d_in[0];
  const int*   ei  = (const int*)d_in[1];
  const float* W1  = (const float*)d_in[2];
  const float* as1 = (const float*)d_in[3];
  const float* ad1 = (const float*)d_in[4];
  const float* b1  = (const float*)d_in[5];
  const float* W2  = (const float*)d_in[6];
  const float* as2 = (const float*)d_in[7];
  const float* ad2 = (const float*)d_in[8];
  const float* b2  = (const float*)d_in[9];
  float* out = (float*)d_out;

  const int N  = in_sizes[0] / 128;     // F_in = 128
  const int E  = in_sizes[1] / 2;       // edge_index is [2, E]
  const int E2 = E + N;                 // + self loops
  const int* srcI = ei;
  const int* dstI = ei + E;

  // workspace layout
  char* ws = (char*)d_ws;
  auto al = [](size_t o) { return (o + 255) & ~(size_t)255; };
  size_t off = 0;
  float*    h1   = (float*)(ws + off);                  off = al(off + (size_t)N * 128 * 4);
  size_t accOff = off;
  float*    acc1 = (float*)(ws + off);                  off = al(off + (size_t)N * 128 * 4);
  float*    ex1  = (float*)(ws + off);                  off = al(off + (size_t)E2 * 2 * 4);
  unsigned* mk1  = (unsigned*)(ws + off);               off = al(off + (size_t)N * 2 * 4);
  float*    den1 = (float*)(ws + off);                  off = al(off + (size_t)N * 2 * 4);
  float*    aS1  = (float*)(ws + off);                  off = al(off + (size_t)N * 2 * 4);
  float*    aD1  = (float*)(ws + off);                  off = al(off + (size_t)N * 2 * 4);
  unsigned* mk2  = (unsigned*)(ws + off);               off = al(off + (size_t)N * 4);
  float*    den2 = (float*)(ws + off);                  off = al(off + (size_t)N * 4);
  float*    aS2  = (float*)(ws + off);                  off = al(off + (size_t)N * 4);
  float*    aD2  = (float*)(ws + off);                  off = al(off + (size_t)N * 4);
  float*    ex2  = (float*)(ws + off);                  off = al(off + (size_t)E2 * 4);
  // layer-2 GEMM output reuses the (then-dead) acc1 region
  float*    h2   = (float*)(ws + accOff);

  // ---------- layer 1 ----------
  fill_u32_kernel<<<nblk((long long)N * 128, 256), 256, 0, stream>>>((uint32_t*)acc1, 0u, (long long)N * 128);
  fill_u32_kernel<<<nblk((long long)N * 2, 256), 256, 0, stream>>>((uint32_t*)mk1, 0u, (long long)N * 2);
  fill_u32_kernel<<<nblk((long long)N * 2, 256), 256, 0, stream>>>((uint32_t*)den1, 0u, (long long)N * 2);

  gemm_f32_wmma<128><<<N / 16, 256, 0, stream>>>(x, W1, h1);
  attn_dots<128, 2><<<nblk(N, 8), 256, 0, stream>>>(h1, as1, ad1, aS1, aD1, N);
  edge_max<2><<<nblk(E2, 256), 256, 0, stream>>>(srcI, dstI, E, E2, aS1, aD1, mk1);
  edge_exp<2><<<nblk(E2, 256), 256, 0, stream>>>(srcI, dstI, E, E2, aS1, aD1, mk1, ex1, den1);
  edge_scatter<128, 2><<<nblk((long long)E2 * 128, 256), 256, 0, stream>>>(
      srcI, dstI, E, E2, h1, ex1, den1, acc1);
  bias_elu_kernel<<<nblk((long long)N * 128, 256), 256, 0, stream>>>(acc1, b1, h1, (long long)N * 128, 128);

  // ---------- layer 2 ----------
  fill_u32_kernel<<<nblk((long long)N * 64, 256), 256, 0, stream>>>((uint32_t*)out, 0u, (long long)N * 64);
  fill_u32_kernel<<<nblk(N, 256), 256, 0, stream>>>((uint32_t*)mk2, 0u, (long long)N);
  fill_u32_kernel<<<nblk(N, 256), 256, 0, stream>>>((uint32_t*)den2, 0u, (long long)N);

  gemm_f32_wmma<64><<<N / 16, 128, 0, stream>>>(h1, W2, h2);
  attn_dots<64, 1><<<nblk(N, 8), 256, 0, stream>>>(h2, as2, ad2, aS2, aD2, N);
  edge_max<1><<<nblk(E2, 256), 256, 0, stream>>>(srcI, dstI, E, E2, aS2, aD2, mk2);
  edge_exp<1><<<nblk(E2, 256), 256, 0, stream>>>(srcI, dstI, E, E2, aS2, aD2, mk2, ex2, den2);
  edge_scatter<64, 1><<<nblk((long long)E2 * 64, 256), 256, 0, stream>>>(
      srcI, dstI, E, E2, h2, ex2, den2, out);
  bias_add_kernel<<<nblk((long long)N * 64, 256), 256, 0, stream>>>(out, b2, (long long)N * 64, 64);
}